// EGNN_dynamics_QM9_6253472383640
// MI455X (gfx1250) — compile-verified
//
#include <hip/hip_runtime.h>
#include <math.h>

// ---------------- problem constants ----------------
constexpr int Bq = 32, Nq = 48, Hq = 128;
constexpr int ROWS = Bq * Nq;          // 1536 node rows
constexpr int EDGES = Bq * Nq * Nq;    // 73728 edges
constexpr int SSTRIDE = 132;           // padded LDS row stride (floats)
constexpr int GROUP = 4;               // (b,i) groups per block in edge kernels

typedef float v2f __attribute__((ext_vector_type(2)));
typedef float v8f __attribute__((ext_vector_type(8)));

// CDNA5 fp32 WMMA: D(16x16) = A(16x4) * B(4x16) + C  -- full fp32 precision
__device__ __forceinline__ v8f wmma4(v2f a, v2f b, v8f c) {
  return __builtin_amdgcn_wmma_f32_16x16x4_f32(false, a, false, b, (short)0, c,
                                               false, false);
}

__device__ __forceinline__ float silu_f(float x) {
  return x / (1.0f + expf(-x));
}

// ---------------------------------------------------------------------------
// h = emb(concat(h_dims*mask, t, context));  x = x0 = xh[...,:3]*mask
// ---------------------------------------------------------------------------
__global__ void k_embed(const float* __restrict__ xh, const float* __restrict__ t,
                        const float* __restrict__ nmask, const float* __restrict__ ctx,
                        const float* __restrict__ W, const float* __restrict__ bias,
                        float* __restrict__ h, float* __restrict__ x,
                        float* __restrict__ x0) {
  int gid = blockIdx.x * 256 + threadIdx.x;
  if (gid >= ROWS * Hq) return;
  int row = gid >> 7, c = gid & 127;
  int b = row / Nq;
  float nm = nmask[row];
  float acc = bias[c];
  #pragma unroll
  for (int k = 0; k < 6; ++k) acc += (xh[row * 9 + 3 + k] * nm) * W[k * Hq + c];
  acc += t[b] * W[6 * Hq + c];
  #pragma unroll
  for (int k = 0; k < 3; ++k) acc += ctx[row * 3 + k] * W[(7 + k) * Hq + c];
  h[row * Hq + c] = acc;
  if (c < 3) {
    float xv = xh[row * 9 + c] * nm;
    x[row * 3 + c] = xv;
    x0[row * 3 + c] = xv;
  }
}

// ---------------------------------------------------------------------------
// radial[b,i,j] = |x_i - x_j|^2 ; cdiff = diff / sqrt(radial + 1e-8) (optional)
// ---------------------------------------------------------------------------
__global__ void k_geom(const float* __restrict__ x, float* __restrict__ radial,
                       float* __restrict__ cdiff) {
  int bi = blockIdx.x;          // b*Nq + i
  int j = threadIdx.x;
  if (j >= Nq) return;
  int b = bi / Nq;
  float xd[3], r = 0.f;
  #pragma unroll
  for (int d = 0; d < 3; ++d) {
    xd[d] = x[bi * 3 + d] - x[(b * Nq + j) * 3 + d];
    r += xd[d] * xd[d];
  }
  radial[bi * Nq + j] = r;
  if (cdiff) {
    float inv = 1.0f / sqrtf(r + 1e-8f);
    #pragma unroll
    for (int d = 0; d < 3; ++d) cdiff[(bi * Nq + j) * 3 + d] = xd[d] * inv;
  }
}

// ---------------------------------------------------------------------------
// Generic row GEMM: C[1536,128] = A[1536,128] @ W[128,128] (+bias)
// One block per 16-row tile; 8 waves = 8 column tiles. fp32 WMMA.
// ---------------------------------------------------------------------------
__global__ __launch_bounds__(256) void k_gemm_rows(const float* __restrict__ A,
                                                   const float* __restrict__ W,
                                                   const float* __restrict__ bias,
                                                   float* __restrict__ C) {
  __shared__ float Ah[16][SSTRIDE];
  const int tid = threadIdx.x;
  const int m0 = blockIdx.x * 16;
  for (int idx = tid; idx < 16 * Hq; idx += 256) {
    int r = idx >> 7, c = idx & 127;
    Ah[r][c] = A[(m0 + r) * Hq + c];
  }
  __syncthreads();
  const int wave = tid >> 5, lane = tid & 31, half = lane >> 4, ln = lane & 15;
  const int n0 = wave * 16;
  v8f acc = {};
  for (int kt = 0; kt < 32; ++kt) {
    int k = 4 * kt + 2 * half;
    v2f a = *(const v2f*)&Ah[ln][k];
    v2f bf;
    bf.x = W[(k + 0) * Hq + n0 + ln];
    bf.y = W[(k + 1) * Hq + n0 + ln];
    acc = wmma4(a, bf, acc);
  }
  float bn = bias ? bias[n0 + ln] : 0.0f;
  #pragma unroll
  for (int v = 0; v < 8; ++v) {
    int row = m0 + v + 8 * half;
    C[row * Hq + n0 + ln] = acc[v] + bn;
  }
}

// ---------------------------------------------------------------------------
// Node update: h = (h + W1^T silu(W0a^T h + W0b^T agg + b0) + b1) * node_mask
// ---------------------------------------------------------------------------
__global__ __launch_bounds__(256) void k_node(float* __restrict__ h,
                                              const float* __restrict__ agg,
                                              const float* __restrict__ W0,
                                              const float* __restrict__ b0,
                                              const float* __restrict__ W1,
                                              const float* __restrict__ b1,
                                              const float* __restrict__ nmask) {
  __shared__ float Ah[16][SSTRIDE];
  __shared__ float Ag[16][SSTRIDE];
  __shared__ float U[16][SSTRIDE];
  const int tid = threadIdx.x;
  const int m0 = blockIdx.x * 16;
  for (int idx = tid; idx < 16 * Hq; idx += 256) {
    int r = idx >> 7, c = idx & 127;
    Ah[r][c] = h[(m0 + r) * Hq + c];
    Ag[r][c] = agg[(m0 + r) * Hq + c];
  }
  __syncthreads();
  const int wave = tid >> 5, lane = tid & 31, half = lane >> 4, ln = lane & 15;
  const int n0 = wave * 16;
  v8f acc = {};
  for (int kt = 0; kt < 32; ++kt) {
    int k = 4 * kt + 2 * half;
    v2f a = *(const v2f*)&Ah[ln][k];
    v2f bf;
    bf.x = W0[(k + 0) * Hq + n0 + ln];
    bf.y = W0[(k + 1) * Hq + n0 + ln];
    acc = wmma4(a, bf, acc);
  }
  for (int kt = 0; kt < 32; ++kt) {
    int k = 4 * kt + 2 * half;
    v2f a = *(const v2f*)&Ag[ln][k];
    v2f bf;
    bf.x = W0[(128 + k) * Hq + n0 + ln];
    bf.y = W0[(129 + k) * Hq + n0 + ln];
    acc = wmma4(a, bf, acc);
  }
  float bb = b0[n0 + ln];
  #pragma unroll
  for (int v = 0; v < 8; ++v) U[v + 8 * half][n0 + ln] = silu_f(acc[v] + bb);
  __syncthreads();
  v8f acc2 = {};
  for (int kt = 0; kt < 32; ++kt) {
    int k = 4 * kt + 2 * half;
    v2f a = *(const v2f*)&U[ln][k];
    v2f bf;
    bf.x = W1[(k + 0) * Hq + n0 + ln];
    bf.y = W1[(k + 1) * Hq + n0 + ln];
    acc2 = wmma4(a, bf, acc2);
  }
  float b1n = b1[n0 + ln];
  #pragma unroll
  for (int v = 0; v < 8; ++v) {
    int row = m0 + v + 8 * half;
    float nm = nmask[row];
    h[row * Hq + n0 + ln] = (Ah[v + 8 * half][n0 + ln] + acc2[v] + b1n) * nm;
  }
}

// ---------------------------------------------------------------------------
// Fused edge pass, GROUP (b,i) groups per block with LDS-staged transposed W1:
// S = silu(HiP[i]+HjP[j]+rad*wr+d0*wd); WMMA S@W1; silu; *edge_mask;
// reduce over j -> agg[b,i,:]/100.   Hot loop is pure LDS + WMMA.
// ---------------------------------------------------------------------------
__global__ __launch_bounds__(256) void k_edge_agg(
    const float* __restrict__ HiP, const float* __restrict__ HjP,
    const float* __restrict__ wr, const float* __restrict__ wd,
    const float* __restrict__ radial, const float* __restrict__ d0,
    const float* __restrict__ emask, const float* __restrict__ W1,
    const float* __restrict__ b1, float* __restrict__ agg) {
  __shared__ float Wt[Hq][SSTRIDE];          // Wt[n][k] = W1[k][n]
  __shared__ float S[Nq][SSTRIDE];
  __shared__ float hip[Hq], vr[Hq], vd[Hq];
  __shared__ float rad[Nq], dd[Nq], em[Nq];
  const int tid = threadIdx.x;
  const int wave = tid >> 5, lane = tid & 31, half = lane >> 4, ln = lane & 15;
  const int n0 = wave * 16;

  for (int idx = tid; idx < Hq * Hq; idx += 256) {  // coalesced global reads
    int k = idx >> 7, n = idx & 127;
    Wt[n][k] = W1[k * Hq + n];
  }
  if (tid < Hq) { vr[tid] = wr[tid]; vd[tid] = wd[tid]; }

  for (int g = 0; g < GROUP; ++g) {
    const int bi = blockIdx.x * GROUP + g;   // b*Nq + i
    const int b = bi / Nq;
    __syncthreads();                         // Wt staged / prev iter done
    if (tid < Hq) hip[tid] = HiP[bi * Hq + tid];
    if (tid < Nq) {
      rad[tid] = radial[bi * Nq + tid];
      dd[tid]  = d0[bi * Nq + tid];
      em[tid]  = emask[bi * Nq + tid];
    }
    __syncthreads();
    for (int idx = tid; idx < Nq * Hq; idx += 256) {
      int j = idx >> 7, n = idx & 127;
      float pre = hip[n] + HjP[(b * Nq + j) * Hq + n] + rad[j] * vr[n] + dd[j] * vd[n];
      S[j][n] = silu_f(pre);
    }
    __syncthreads();
    v8f a0 = {}, a1 = {}, a2 = {};
    for (int kt = 0; kt < 32; ++kt) {
      int k = 4 * kt + 2 * half;
      v2f bf = *(const v2f*)&Wt[n0 + ln][k];
      v2f aa;
      aa = *(const v2f*)&S[ln][k];      a0 = wmma4(aa, bf, a0);
      aa = *(const v2f*)&S[16 + ln][k]; a1 = wmma4(aa, bf, a1);
      aa = *(const v2f*)&S[32 + ln][k]; a2 = wmma4(aa, bf, a2);
    }
    float bb = b1[n0 + ln];
    float P = 0.f;
    #pragma unroll
    for (int v = 0; v < 8; ++v) {
      int r0 = v + 8 * half;
      P += silu_f(a0[v] + bb) * em[r0];
      P += silu_f(a1[v] + bb) * em[16 + r0];
      P += silu_f(a2[v] + bb) * em[32 + r0];
    }
    P += __shfl_xor(P, 16, 32);
    if (lane < 16) agg[bi * Hq + n0 + ln] = P * 0.01f;
  }
}

// ---------------------------------------------------------------------------
// Fused equivariant pass (same staging): e2 = silu(c1(silu(pre)));
// phi = e2 @ w_last;  x[b,i] += sum_j cdiff*phi*edge_mask / 100
// ---------------------------------------------------------------------------
__global__ __launch_bounds__(256) void k_equiv_edge(
    const float* __restrict__ HiP, const float* __restrict__ HjP,
    const float* __restrict__ wr, const float* __restrict__ wd,
    const float* __restrict__ radial, const float* __restrict__ d0,
    const float* __restrict__ emask, const float* __restrict__ W1,
    const float* __restrict__ b1, const float* __restrict__ wlast,
    const float* __restrict__ cdiff, float* __restrict__ x) {
  __shared__ float Wt[Hq][SSTRIDE];
  __shared__ float S[Nq][SSTRIDE];
  __shared__ float hip[Hq], vr[Hq], vd[Hq], wl[Hq];
  __shared__ float rad[Nq], dd[Nq], em[Nq], phi[Nq];
  __shared__ float xacc[3];
  const int tid = threadIdx.x;
  const int wave = tid >> 5, lane = tid & 31, half = lane >> 4, ln = lane & 15;
  const int n0 = wave * 16;

  for (int idx = tid; idx < Hq * Hq; idx += 256) {
    int k = idx >> 7, n = idx & 127;
    Wt[n][k] = W1[k * Hq + n];
  }
  if (tid < Hq) { vr[tid] = wr[tid]; vd[tid] = wd[tid]; wl[tid] = wlast[tid]; }

  for (int g = 0; g < GROUP; ++g) {
    const int bi = blockIdx.x * GROUP + g;
    const int b = bi / Nq;
    __syncthreads();
    if (tid < Hq) hip[tid] = HiP[bi * Hq + tid];
    if (tid < Nq) {
      rad[tid] = radial[bi * Nq + tid];
      dd[tid]  = d0[bi * Nq + tid];
      em[tid]  = emask[bi * Nq + tid];
      phi[tid] = 0.f;
    }
    if (tid < 3) xacc[tid] = 0.f;
    __syncthreads();
    for (int idx = tid; idx < Nq * Hq; idx += 256) {
      int j = idx >> 7, n = idx & 127;
      float pre = hip[n] + HjP[(b * Nq + j) * Hq + n] + rad[j] * vr[n] + dd[j] * vd[n];
      S[j][n] = silu_f(pre);
    }
    __syncthreads();
    v8f a0 = {}, a1 = {}, a2 = {};
    for (int kt = 0; kt < 32; ++kt) {
      int k = 4 * kt + 2 * half;
      v2f bf = *(const v2f*)&Wt[n0 + ln][k];
      v2f aa;
      aa = *(const v2f*)&S[ln][k];      a0 = wmma4(aa, bf, a0);
      aa = *(const v2f*)&S[16 + ln][k]; a1 = wmma4(aa, bf, a1);
      aa = *(const v2f*)&S[32 + ln][k]; a2 = wmma4(aa, bf, a2);
    }
    float bb = b1[n0 + ln], wln = wl[n0 + ln];
    v8f t0, t1, t2;
    #pragma unroll
    for (int v = 0; v < 8; ++v) {
      t0[v] = silu_f(a0[v] + bb) * wln;
      t1[v] = silu_f(a1[v] + bb) * wln;
      t2[v] = silu_f(a2[v] + bb) * wln;
    }
    #pragma unroll
    for (int off = 1; off < 16; off <<= 1) {
      #pragma unroll
      for (int v = 0; v < 8; ++v) {
        t0[v] += __shfl_xor(t0[v], off, 32);
        t1[v] += __shfl_xor(t1[v], off, 32);
        t2[v] += __shfl_xor(t2[v], off, 32);
      }
    }
    if (ln == 0) {
      #pragma unroll
      for (int v = 0; v < 8; ++v) {
        atomicAdd(&phi[v + 8 * half], t0[v]);
        atomicAdd(&phi[16 + v + 8 * half], t1[v]);
        atomicAdd(&phi[32 + v + 8 * half], t2[v]);
      }
    }
    __syncthreads();
    if (tid < Nq) {
      float wgt = phi[tid] * em[tid];
      const float* cdp = &cdiff[(bi * Nq + tid) * 3];
      atomicAdd(&xacc[0], cdp[0] * wgt);
      atomicAdd(&xacc[1], cdp[1] * wgt);
      atomicAdd(&xacc[2], cdp[2] * wgt);
    }
    __syncthreads();
    if (tid < 3) x[bi * 3 + tid] += xacc[tid] * 0.01f;
  }
}

// ---------------------------------------------------------------------------
// Output: vel = (x-x0)*mask, subtract masked mean; h_out = emb_out(h)[:,:6]*mask
// ---------------------------------------------------------------------------
__global__ __launch_bounds__(256) void k_final(
    const float* __restrict__ h, const float* __restrict__ x,
    const float* __restrict__ x0, const float* __restrict__ nmask,
    const float* __restrict__ Wo, const float* __restrict__ bo,
    float* __restrict__ out) {
  __shared__ float vel[Nq][3];
  __shared__ float vsum[3];
  __shared__ float msum;
  int b = blockIdx.x, tid = threadIdx.x;
  if (tid < 3) vsum[tid] = 0.f;
  if (tid == 0) msum = 0.f;
  __syncthreads();
  if (tid < Nq) atomicAdd(&msum, nmask[b * Nq + tid]);
  if (tid < Nq * 3) {
    int n = tid / 3, d = tid % 3, row = b * Nq + n;
    float v = (x[row * 3 + d] - x0[row * 3 + d]) * nmask[row];
    vel[n][d] = v;
    atomicAdd(&vsum[d], v);
  }
  __syncthreads();
  if (tid < Nq * 3) {
    int n = tid / 3, d = tid % 3, row = b * Nq + n;
    out[row * 9 + d] = vel[n][d] - vsum[d] / msum * nmask[row];
  }
  for (int oi = tid; oi < Nq * 6; oi += 256) {
    int n = oi / 6, c = oi % 6, row = b * Nq + n;
    float acc = bo[c];
    for (int k = 0; k < Hq; ++k) acc += h[row * Hq + k] * Wo[k * 10 + c];
    out[row * 9 + 3 + c] = acc * nmask[row];
  }
}

// ---------------------------------------------------------------------------
extern "C" void kernel_launch(void* const* d_in, const int* in_sizes, int n_in,
                              void* d_out, int out_size, void* d_ws, size_t ws_size,
                              hipStream_t stream) {
  (void)in_sizes; (void)n_in; (void)out_size; (void)ws_size;
  const float* t     = (const float*)d_in[0];
  const float* xh    = (const float*)d_in[1];
  const float* nmask = (const float*)d_in[2];
  const float* emask = (const float*)d_in[3];
  const float* ctx   = (const float*)d_in[4];

  // JAX tree_flatten order (dict keys sorted): blocks, emb, emb_out.
  // per block: equiv{c0,c1,w_last}, gcls[0..1]{edge0,edge1,node0,node1}
  struct Gcl { const float *e0W, *e0b, *e1W, *e1b, *n0W, *n0b, *n1W, *n1b; };
  struct Blk { const float *c0W, *c0b, *c1W, *c1b, *wl; Gcl g[2]; };
  Blk blk[4];
  int p = 5;
  for (int l = 0; l < 4; ++l) {
    blk[l].c0W = (const float*)d_in[p++]; blk[l].c0b = (const float*)d_in[p++];
    blk[l].c1W = (const float*)d_in[p++]; blk[l].c1b = (const float*)d_in[p++];
    blk[l].wl  = (const float*)d_in[p++];
    for (int g = 0; g < 2; ++g) {
      blk[l].g[g].e0W = (const float*)d_in[p++]; blk[l].g[g].e0b = (const float*)d_in[p++];
      blk[l].g[g].e1W = (const float*)d_in[p++]; blk[l].g[g].e1b = (const float*)d_in[p++];
      blk[l].g[g].n0W = (const float*)d_in[p++]; blk[l].g[g].n0b = (const float*)d_in[p++];
      blk[l].g[g].n1W = (const float*)d_in[p++]; blk[l].g[g].n1b = (const float*)d_in[p++];
    }
  }
  const float* embW  = (const float*)d_in[p++];
  const float* embB  = (const float*)d_in[p++];
  const float* eoW   = (const float*)d_in[p++];
  const float* eoB   = (const float*)d_in[p++];

  float* ws  = (float*)d_ws;
  float* h   = ws; ws += ROWS * Hq;
  float* x   = ws; ws += ROWS * 3;
  float* x0  = ws; ws += ROWS * 3;
  float* d0v = ws; ws += EDGES;
  float* rad = ws; ws += EDGES;
  float* cd  = ws; ws += EDGES * 3;
  float* HiP = ws; ws += ROWS * Hq;
  float* HjP = ws; ws += ROWS * Hq;
  float* agg = ws; ws += ROWS * Hq;

  k_embed<<<(ROWS * Hq + 255) / 256, 256, 0, stream>>>(xh, t, nmask, ctx, embW, embB,
                                                       h, x, x0);
  k_geom<<<ROWS, 64, 0, stream>>>(x0, d0v, nullptr);   // d0 from initial masked x

  for (int l = 0; l < 4; ++l) {
    k_geom<<<ROWS, 64, 0, stream>>>(x, rad, cd);
    for (int g = 0; g < 2; ++g) {
      const Gcl& G = blk[l].g[g];
      k_gemm_rows<<<ROWS / 16, 256, 0, stream>>>(h, G.e0W, G.e0b, HiP);
      k_gemm_rows<<<ROWS / 16, 256, 0, stream>>>(h, G.e0W + 128 * Hq, nullptr, HjP);
      k_edge_agg<<<ROWS / GROUP, 256, 0, stream>>>(HiP, HjP, G.e0W + 256 * Hq,
                                                   G.e0W + 257 * Hq, rad, d0v, emask,
                                                   G.e1W, G.e1b, agg);
      k_node<<<ROWS / 16, 256, 0, stream>>>(h, agg, G.n0W, G.n0b, G.n1W, G.n1b, nmask);
    }
    const Blk& Bk = blk[l];
    k_gemm_rows<<<ROWS / 16, 256, 0, stream>>>(h, Bk.c0W, Bk.c0b, HiP);
    k_gemm_rows<<<ROWS / 16, 256, 0, stream>>>(h, Bk.c0W + 128 * Hq, nullptr, HjP);
    k_equiv_edge<<<ROWS / GROUP, 256, 0, stream>>>(HiP, HjP, Bk.c0W + 256 * Hq,
                                                   Bk.c0W + 257 * Hq, rad, d0v, emask,
                                                   Bk.c1W, Bk.c1b, Bk.wl, cd, x);
  }
  k_final<<<Bq, 256, 0, stream>>>(h, x, x0, nmask, eoW, eoB, (float*)d_out);
}